// GazeLSTM_68667937128935
// MI455X (gfx1250) — compile-verified
//
#include <hip/hip_runtime.h>
#include <hip/hip_bf16.h>
#include <stdint.h>

// ---------------------------------------------------------------------------
// GazeLSTM forward, MI455X (gfx1250).
// Memory-bound (~200MB traffic -> ~8.6us @ 23.3TB/s). R_mode is staged into
// LDS with the CDNA5 Tensor Data Mover (tensor_load_to_lds, th:TH_LOAD_NT),
// triple-buffered per wave (prefetch distance 2 to exceed the HBM
// bandwidth-delay product). Compute is per-thread 3x3 chains + a 33-step
// tiny LSTM using the gfx1250 hardware tanh; LSTM inputs are re-read from L2
// in 8-row register batches (independent back-to-back loads) to avoid
// serially-dependent load latency while staying I$-friendly.
// ---------------------------------------------------------------------------

typedef uint32_t u32x4 __attribute__((ext_vector_type(4)));
typedef uint32_t u32x8 __attribute__((ext_vector_type(8)));

#define BATCH   65536
#define ROWF    567          // floats per element in R_mode (63*9)
#define TILE_W  63           // floats per TDM tile row (7 timesteps)
#define TILE_H  32           // tile rows = elements per wave
#define BUF_F   (TILE_W * TILE_H)   // 2016 floats per buffer
#define NBUF    3            // triple buffer

// ---- activations: prefer gfx1250 V_TANH_F32 -------------------------------
#if defined(__has_builtin)
# if __has_builtin(__builtin_amdgcn_tanhf)
#  define HW_TANH(x) __builtin_amdgcn_tanhf(x)
# elif __has_builtin(__builtin_amdgcn_tanh_f32)
#  define HW_TANH(x) __builtin_amdgcn_tanh_f32(x)
# endif
#endif

__device__ __forceinline__ float tanhx(float x) {
#ifdef HW_TANH
    return HW_TANH(x);
#else
    return 1.0f - 2.0f / (__expf(2.0f * x) + 1.0f);
#endif
}
__device__ __forceinline__ float sigm(float x) {
#ifdef HW_TANH
    return __builtin_fmaf(0.5f, tanhx(0.5f * x), 0.5f);
#else
    return 1.0f / (1.0f + __expf(-x));
#endif
}

// Issue one TDM 2D tile load: TILE_H rows x TILE_W floats, row stride ROWF.
// All descriptor fields must be wave-uniform (SGPR groups). R_mode is
// streamed once -> non-temporal hint keeps L2 free for the output rows the
// LSTM re-reads.
__device__ __forceinline__ void tdm_load(uint32_t lds_byte_addr, const float* gsrc) {
    uint64_t ga = (uint64_t)(uintptr_t)gsrc;
    u32x4 g0;
    g0[0] = 1u;                                   // count=1, user descriptor
    g0[1] = lds_byte_addr;                        // LDS destination (bytes)
    g0[2] = (uint32_t)ga;                         // global_addr[31:0]
    g0[3] = ((uint32_t)(ga >> 32) & 0x01FFFFFFu)  // global_addr[56:32]
            | (2u << 30);                         // type=2 ("image")
    u32x8 g1;
    g1[0] = 2u << 16;                             // data_size=2 -> 4-byte elems
    g1[1] = 0u;                                   // tensor_dim0[15:0] (dim0 = 1<<20)
    g1[2] = 0x10u;                                // tensor_dim0[31:16]=0x10, dim1 lo=0
    g1[3] = 0x10u | ((uint32_t)TILE_W << 16);     // tensor_dim1 hi=0x10, tile_dim0=63
    g1[4] = (uint32_t)TILE_H;                     // tile_dim1=32, tile_dim2=0
    g1[5] = (uint32_t)ROWF;                       // tensor_dim0_stride lo (567)
    g1[6] = 0u;                                   // stride0 hi / stride1 lo
    g1[7] = 0u;                                   // stride1 hi
    asm volatile("tensor_load_to_lds %0, %1 th:TH_LOAD_NT" :: "s"(g0), "s"(g1) : "memory");
}

// chunk start (R index) for chunk c; chunks 0..4 backward (t=31..0),
// chunks 5..9 forward (t=32..62). All tiles are 7 timesteps (63 floats) wide.
__device__ __forceinline__ int chunk_start(int c) {
    if (c < 5) { int tl = 31 - 7 * c - 6; return tl < 0 ? 0 : tl; }
    int lo = 32 + 7 * (c - 5);
    return lo > 56 ? 56 : lo;
}

__global__ __launch_bounds__(64) void gaze_kernel(
    const float* __restrict__ dirp,
    const float* __restrict__ R,
    const float* __restrict__ wih_g,
    const float* __restrict__ whh_g,
    const float* __restrict__ bih_g,
    const float* __restrict__ bhh_g,
    float* out)
{
    __shared__ __align__(16) float smem[2][NBUF][BUF_F];   // [wave][buffer][tile]

    const int tid  = threadIdx.x;
    const int lane = tid & 31;
    const int wv   = tid >> 5;
    const int e    = blockIdx.x * 64 + tid;

    // Wave-uniform values for the TDM descriptor.
    uint32_t e0  = (uint32_t)__builtin_amdgcn_readfirstlane((int)(blockIdx.x * 64 + (wv << 5)));
    uint32_t wvu = (uint32_t)__builtin_amdgcn_readfirstlane(wv);
    uint32_t lds_base = (uint32_t)(uintptr_t)(&smem[0][0][0]) + wvu * (NBUF * BUF_F * 4u);
    const float* gbase = R + (uint64_t)e0 * ROWF;

    // Prime the pipeline: chunks 0 and 1.
    tdm_load(lds_base + 0u * (BUF_F * 4u), gbase + chunk_start(0) * 9);
    tdm_load(lds_base + 1u * (BUF_F * 4u), gbase + chunk_start(1) * 9);

    const float d0 = dirp[e * 3 + 0];
    const float d1 = dirp[e * 3 + 1];
    const float d2 = dirp[e * 3 + 2];
    float Q[9] = {1,0,0, 0,1,0, 0,0,1};   // backward accumulator (R^T chain)
    float P[9] = {1,0,0, 0,1,0, 0,0,1};   // forward accumulator
    float* orow = out + (size_t)e * 192;

    for (int c = 0; c < 10; ++c) {
        if (c + 2 < 10) {
            tdm_load(lds_base + (uint32_t)((c + 2) % NBUF) * (BUF_F * 4u),
                     gbase + chunk_start(c + 2) * 9);
            asm volatile("s_wait_tensorcnt 0x2" ::: "memory");
        } else if (c + 1 < 10) {
            asm volatile("s_wait_tensorcnt 0x1" ::: "memory");
        } else {
            asm volatile("s_wait_tensorcnt 0x0" ::: "memory");
        }
        const float* rb = &smem[wv][c % NBUF][lane * TILE_W];
        const int st = chunk_start(c);

        if (c < 5) {
            // Backward: t descending; Q <- Q * R[t]^T ; dirs[t] = Q * dir
            const int thi = 31 - 7 * c;
            const int n   = (c == 4) ? 4 : 7;
            for (int k = 0; k < n; ++k) {
                const int t = thi - k;
                const float* Rt = rb + (t - st) * 9;
                float r[9];
                #pragma unroll
                for (int q = 0; q < 9; ++q) r[q] = Rt[q];
                float Qn[9];
                #pragma unroll
                for (int i = 0; i < 3; ++i)
                    #pragma unroll
                    for (int j = 0; j < 3; ++j)
                        Qn[i*3+j] = Q[i*3+0]*r[j*3+0] + Q[i*3+1]*r[j*3+1] + Q[i*3+2]*r[j*3+2];
                #pragma unroll
                for (int q = 0; q < 9; ++q) Q[q] = Qn[q];
                // rows 0..31 are re-read by the LSTM -> keep regular-temporal
                orow[t*3+0] = Q[0]*d0 + Q[1]*d1 + Q[2]*d2;
                orow[t*3+1] = Q[3]*d0 + Q[4]*d1 + Q[5]*d2;
                orow[t*3+2] = Q[6]*d0 + Q[7]*d1 + Q[8]*d2;
            }
        } else {
            // Forward: t ascending; P <- P * R[t] ; dirs[t+1] = P * dir
            const int f   = c - 5;
            const int plo = 32 + 7 * f;
            const int phi = (plo + 6 > 62) ? 62 : plo + 6;
            for (int t = plo; t <= phi; ++t) {
                const float* Rt = rb + (t - st) * 9;
                float r[9];
                #pragma unroll
                for (int q = 0; q < 9; ++q) r[q] = Rt[q];
                float Pn[9];
                #pragma unroll
                for (int i = 0; i < 3; ++i)
                    #pragma unroll
                    for (int j = 0; j < 3; ++j)
                        Pn[i*3+j] = P[i*3+0]*r[0*3+j] + P[i*3+1]*r[1*3+j] + P[i*3+2]*r[2*3+j];
                #pragma unroll
                for (int q = 0; q < 9; ++q) P[q] = Pn[q];
                // rows 33..63 are never re-read -> non-temporal stores
                __builtin_nontemporal_store(P[0]*d0 + P[1]*d1 + P[2]*d2, &orow[(t+1)*3+0]);
                __builtin_nontemporal_store(P[3]*d0 + P[4]*d1 + P[5]*d2, &orow[(t+1)*3+1]);
                __builtin_nontemporal_store(P[6]*d0 + P[7]*d1 + P[8]*d2, &orow[(t+1)*3+2]);
            }
        }
    }

    // ---- tiny LSTM, only steps 0..32 matter (hs[32] replaces row 32) ----
    float wih[12][3], whh[12][3], bb[12];
    #pragma unroll
    for (int g = 0; g < 12; ++g) {
        #pragma unroll
        for (int j = 0; j < 3; ++j) {
            wih[g][j] = wih_g[g * 3 + j];
            whh[g][j] = whh_g[g * 3 + j];
        }
        bb[g] = bih_g[g] + bhh_g[g];
    }

    float h0 = 0.f, h1 = 0.f, h2 = 0.f;
    float c0 = 0.f, c1 = 0.f, c2 = 0.f;

    auto lstm_step = [&](float x0, float x1, float x2) {
        float gate[12];
        #pragma unroll
        for (int g = 0; g < 12; ++g)
            gate[g] = bb[g] + wih[g][0]*x0 + wih[g][1]*x1 + wih[g][2]*x2
                            + whh[g][0]*h0 + whh[g][1]*h1 + whh[g][2]*h2;
        const float i0 = sigm(gate[0]), i1 = sigm(gate[1]), i2 = sigm(gate[2]);
        const float f0 = sigm(gate[3]), f1 = sigm(gate[4]), f2 = sigm(gate[5]);
        const float g0 = tanhx(gate[6]), g1 = tanhx(gate[7]), g2 = tanhx(gate[8]);
        const float o0 = sigm(gate[9]), o1 = sigm(gate[10]), o2 = sigm(gate[11]);
        c0 = f0 * c0 + i0 * g0;
        c1 = f1 * c1 + i1 * g1;
        c2 = f2 * c2 + i2 * g2;
        h0 = o0 * tanhx(c0);
        h1 = o1 * tanhx(c1);
        h2 = o2 * tanhx(c2);
    };

    // Steps 0..31: batch x-row loads 8 at a time so the L2 reads of a batch
    // issue back-to-back (one latency exposure per 8 steps) while keeping the
    // unrolled body small enough to stay I$-resident.
    for (int tb = 0; tb < 32; tb += 8) {
        float xr[8][3];
        #pragma unroll
        for (int k = 0; k < 8; ++k) {
            xr[k][0] = orow[(tb + k)*3 + 0];
            xr[k][1] = orow[(tb + k)*3 + 1];
            xr[k][2] = orow[(tb + k)*3 + 2];
        }
        #pragma unroll
        for (int k = 0; k < 8; ++k)
            lstm_step(xr[k][0], xr[k][1], xr[k][2]);
    }
    lstm_step(d0, d1, d2);   // step 32, x = dir

    orow[96] = h0;   // row 32 <- hs[32]
    orow[97] = h1;
    orow[98] = h2;
}

extern "C" void kernel_launch(void* const* d_in, const int* in_sizes, int n_in,
                              void* d_out, int out_size, void* d_ws, size_t ws_size,
                              hipStream_t stream) {
    (void)in_sizes; (void)n_in; (void)out_size; (void)d_ws; (void)ws_size;
    const float* dirp = (const float*)d_in[0];
    const float* R    = (const float*)d_in[1];
    // d_in[2] = S_diag, unused by the reference forward.
    const float* wih  = (const float*)d_in[3];
    const float* whh  = (const float*)d_in[4];
    const float* bih  = (const float*)d_in[5];
    const float* bhh  = (const float*)d_in[6];
    float* out = (float*)d_out;

    dim3 grid(BATCH / 64), block(64);
    gaze_kernel<<<grid, block, 0, stream>>>(dirp, R, wih, whh, bih, bhh, out);
}